// SmallestGCNConvNet_16561393893734
// MI455X (gfx1250) — compile-verified
//
#include <hip/hip_runtime.h>

typedef __attribute__((ext_vector_type(2))) float v2f;
typedef __attribute__((ext_vector_type(8))) float v8f;

// ---------------------------------------------------------------------------
// Degree / normalization precompute
// ---------------------------------------------------------------------------
__global__ void init_deg_kernel(float* __restrict__ deg, int n) {
  int i = blockIdx.x * blockDim.x + threadIdx.x;
  if (i < n) deg[i] = 1.0f;  // self-loop contributes 1 to every node's degree
}

__global__ void count_deg_kernel(const int* __restrict__ col, float* __restrict__ deg, int e) {
  int i = blockIdx.x * blockDim.x + threadIdx.x;
  if (i < e) atomicAdd(&deg[col[i]], 1.0f);
}

__global__ void rsqrt_kernel(float* __restrict__ deg, int n) {
  int i = blockIdx.x * blockDim.x + threadIdx.x;
  if (i < n) deg[i] = rsqrtf(deg[i]);  // deg >= 1 always (self-loops)
}

__global__ void zero_kernel(float* __restrict__ p, long n) {
  long i = blockIdx.x * (long)blockDim.x + threadIdx.x;
  if (i < n) p[i] = 0.0f;
}

// ---------------------------------------------------------------------------
// Dense transform T = H @ W using V_WMMA_F32_16X16X4_F32 (full fp32 precision)
// One wave computes one 16x16 output tile; K advances 4 per WMMA.
//
// W is staged in LDS once per block (<= 8 KB). Operand fetches use clamped
// addresses + 0/1 mask multiplies so they lower to unconditional loads with
// v_cndmask, never EXEC-toggling branches: EXEC stays all-1s at the WMMA.
//
// ISA layouts (CDNA5 ISA §7.12.2):
//   A 16x4  f32: lane(0-15)->M=lane, VGPR0=K0, VGPR1=K1; lanes 16-31: K2/K3.
//   B 4x16  f32: lane(0-15)->N=lane, VGPR0=K0, VGPR1=K1; lanes 16-31: K2/K3.
//   C/D 16x16  : VGPR v holds M=v (lanes 0-15) / M=v+8 (lanes 16-31), N=lane&15.
// ---------------------------------------------------------------------------
__global__ void gemm16_wmma_kernel(const float* __restrict__ A,   // [nrows, fi]
                                   const float* __restrict__ W,   // [fi, fo]
                                   float* __restrict__ T,         // [nrows, fo]
                                   int nrows, int fi, int fo, int ntile) {
  __shared__ float wlds[2048];  // max fi*fo = 40*50 = 2000 floats

  // Cooperative LDS fill of W BEFORE any wave can exit (barrier safety).
  const int wcount = fi * fo;
  for (int i = threadIdx.x; i < wcount; i += blockDim.x) wlds[i] = W[i];
  __syncthreads();

  int wave = (int)((blockIdx.x * (long)blockDim.x + threadIdx.x) >> 5);
  int lane = threadIdx.x & 31;
  int mt = wave / ntile;
  int nt = wave - mt * ntile;
  int mtiles = (nrows + 15) >> 4;
  if (mt >= mtiles) return;  // whole-wave uniform exit after the barrier

  const int m0 = mt << 4, n0 = nt << 4;
  const int lm = lane & 15;
  const int hi = lane >> 4;          // which K-half this lane supplies
  const int m  = m0 + lm;            // A row for this lane
  const int nn = n0 + lm;            // B/D column for this lane

  // Clamped (always-legal) coordinates + validity masks
  const int   mc    = m < nrows ? m : nrows - 1;
  const float amrow = m < nrows ? 1.0f : 0.0f;
  const int   nc    = nn < fo ? nn : fo - 1;
  const float bmcol = nn < fo ? 1.0f : 0.0f;
  const float* __restrict__ arow = A + (size_t)mc * fi;

  v8f acc = {};
  for (int k = 0; k < fi; k += 4) {
    const int ka  = k + (hi << 1);
    const int ka0 = ka < fi ? ka : 0;
    const int ka1 = (ka + 1) < fi ? (ka + 1) : 0;
    const float mk0 = (ka     < fi) ? 1.0f : 0.0f;
    const float mk1 = (ka + 1 < fi) ? 1.0f : 0.0f;

    v2f a, b;
    a.x = arow[ka0] * (amrow * mk0);           // unconditional global_load_b32
    a.y = arow[ka1] * (amrow * mk1);
    b.x = wlds[ka0 * fo + nc] * (bmcol * mk0); // unconditional ds_load_b32
    b.y = wlds[ka1 * fo + nc] * (bmcol * mk1);

#if __has_builtin(__builtin_amdgcn_wmma_f32_16x16x4_f32)
    acc = __builtin_amdgcn_wmma_f32_16x16x4_f32(
        /*neg_a=*/false, a, /*neg_b=*/false, b,
        /*c_mod=*/(short)0, acc, /*reuse_a=*/false, /*reuse_b=*/false);
#else
    acc[0] += a.x * b.x + a.y * b.y;  // host-pass / missing-builtin placeholder
#endif
  }

  const int mbase = m0 + (hi << 3);
  for (int v = 0; v < 8; ++v) {
    int mm = mbase + v;
    if (mm < nrows && nn < fo) T[(size_t)mm * fo + nn] = acc[v];
  }
}

// ---------------------------------------------------------------------------
// Edge scatter: agg[col] += t[row] * dinv[row]*dinv[col]
// 2^tpeShift lanes cooperate on one edge; consecutive features ->
// consecutive 4B atomics within the wave (conflict-free, L2-coalesced).
// ---------------------------------------------------------------------------
__global__ void edge_scatter_kernel(const float* __restrict__ t,
                                    const int* __restrict__ row,
                                    const int* __restrict__ col,
                                    const float* __restrict__ dinv,
                                    float* __restrict__ agg,
                                    int e, int fo, int tpeShift) {
  long gid = blockIdx.x * (long)blockDim.x + threadIdx.x;
  long eid = gid >> tpeShift;
  if (eid >= e) return;
  int f0 = (int)(gid & ((1 << tpeShift) - 1));
  int r = row[eid];
  int c = col[eid];
  float w = dinv[r] * dinv[c];
  const float* ts = t + (size_t)r * fo;
  float* as = agg + (size_t)c * fo;
  for (int f = f0; f < fo; f += (1 << tpeShift)) {
    atomicAdd(&as[f], ts[f] * w);
  }
}

// ---------------------------------------------------------------------------
// Finalize: add self-loop term + bias, then ELU + eval-mode BatchNorm.
// Also serves as the grid-wide sync point after the scatter.
// ---------------------------------------------------------------------------
__global__ void finalize_kernel(const float* __restrict__ agg,
                                const float* __restrict__ t,
                                const float* __restrict__ dinv,
                                const float* __restrict__ bias,
                                const float* __restrict__ gamma,
                                const float* __restrict__ beta,
                                const float* __restrict__ mean,
                                const float* __restrict__ var,
                                float* __restrict__ out,
                                int nrows, int fo, int doAct) {
  long gid = blockIdx.x * (long)blockDim.x + threadIdx.x;
  long total = (long)nrows * fo;
  if (gid >= total) return;
  int i = (int)(gid / fo);
  int f = (int)(gid - (long)i * fo);
  float di = dinv[i];
  float v = agg[gid] + t[gid] * di * di + bias[f];
  if (doAct) {
    v = v > 0.0f ? v : (__expf(v) - 1.0f);                            // ELU
    v = (v - mean[f]) * rsqrtf(var[f] + 1e-5f) * gamma[f] + beta[f];  // BN eval
  }
  out[gid] = v;
}

// ---------------------------------------------------------------------------
// Host launcher
// ---------------------------------------------------------------------------
extern "C" void kernel_launch(void* const* d_in, const int* in_sizes, int n_in,
                              void* d_out, int out_size, void* d_ws, size_t ws_size,
                              hipStream_t stream) {
  static const int INS[12]  = {32, 40, 30, 20, 10, 5, 1, 5, 10, 20, 30, 40};
  static const int OUTS[12] = {40, 30, 20, 10, 5, 1, 5, 10, 20, 30, 40, 50};

  const float* x  = (const float*)d_in[0];
  const int*   ei = (const int*)d_in[1];
  const float *Ws[12], *bs[12], *gm[11], *bt[11], *mn[11], *vr[11];
  int idx = 2;
  for (int l = 0; l < 12; ++l) Ws[l] = (const float*)d_in[idx++];
  for (int l = 0; l < 12; ++l) bs[l] = (const float*)d_in[idx++];
  for (int l = 0; l < 11; ++l) gm[l] = (const float*)d_in[idx++];
  for (int l = 0; l < 11; ++l) bt[l] = (const float*)d_in[idx++];
  for (int l = 0; l < 11; ++l) mn[l] = (const float*)d_in[idx++];
  for (int l = 0; l < 11; ++l) vr[l] = (const float*)d_in[idx++];

  const int N = in_sizes[0] / 32;  // x is [N, 32]
  const int E = in_sizes[1] / 2;   // edge_index is [2, E]
  const int* row = ei;             // sources
  const int* col = ei + E;         // targets

  // Workspace layout (fp32): dinv | bufA | bufB | t | agg
  const size_t FMAX = 50;
  float* wsp  = (float*)d_ws;
  float* dinv = wsp;                wsp += N;
  float* bufA = wsp;                wsp += (size_t)N * FMAX;
  float* bufB = wsp;                wsp += (size_t)N * FMAX;
  float* tbuf = wsp;                wsp += (size_t)N * FMAX;
  float* agg  = wsp;

  const int B = 256;  // 8 waves of 32 on CDNA5

  init_deg_kernel<<<(N + B - 1) / B, B, 0, stream>>>(dinv, N);
  count_deg_kernel<<<(E + B - 1) / B, B, 0, stream>>>(col, dinv, E);
  rsqrt_kernel<<<(N + B - 1) / B, B, 0, stream>>>(dinv, N);

  const float* hin = x;
  for (int l = 0; l < 12; ++l) {
    const int fi = INS[l], fo = OUTS[l];
    float* hout = (l == 11) ? (float*)d_out : ((l & 1) ? bufB : bufA);

    // 1) T = H @ W via f32 WMMA (W staged in LDS)
    const int ntile = (fo + 15) / 16;
    const long waves = (long)((N + 15) / 16) * ntile;
    const long gth = waves * 32;
    gemm16_wmma_kernel<<<(int)((gth + B - 1) / B), B, 0, stream>>>(
        hin, Ws[l], tbuf, N, fi, fo, ntile);

    // 2) zero the aggregation buffer
    const long cnt = (long)N * fo;
    zero_kernel<<<(int)((cnt + B - 1) / B), B, 0, stream>>>(agg, cnt);

    // 3) edge scatter-add (lanes-per-edge sized to fo)
    const int tsh = fo >= 32 ? 5 : fo >= 16 ? 4 : fo >= 8 ? 3 : fo >= 4 ? 2 : fo >= 2 ? 1 : 0;
    const long sth = ((long)E) << tsh;
    edge_scatter_kernel<<<(int)((sth + B - 1) / B), B, 0, stream>>>(
        tbuf, row, col, dinv, agg, E, fo, tsh);

    // 4) self-loop + bias + ELU + BN (eval)
    finalize_kernel<<<(int)((cnt + B - 1) / B), B, 0, stream>>>(
        agg, tbuf, dinv, bs[l],
        l < 11 ? gm[l] : bs[l], l < 11 ? bt[l] : bs[l],
        l < 11 ? mn[l] : bs[l], l < 11 ? vr[l] : bs[l],
        hout, N, fo, l < 11 ? 1 : 0);

    hin = hout;
  }
  (void)n_in; (void)out_size; (void)ws_size;
}